// QuanvolutionFilter_65481071406431
// MI455X (gfx1250) — compile-verified
//
#include <hip/hip_runtime.h>
#include <hip/hip_bf16.h>

typedef float v2f __attribute__((ext_vector_type(2)));
typedef float v8f __attribute__((ext_vector_type(8)));

#define N_WIRES 4
#define DIM 16
#define BATCH 4096
#define PSIDE 14
#define NPATCH (PSIDE * PSIDE)          // 196
#define NSTATES (BATCH * NPATCH)        // 802816
#define NTILES (NSTATES / 16)           // 50176
#define WAVES_PER_BLOCK 8
#define TILES_PER_WAVE 4
#define NBLOCKS (NTILES / (WAVES_PER_BLOCK * TILES_PER_WAVE)) // 1568

// ---------------------------------------------------------------------------
// Kernel 1: build the 16x16 complex ansatz unitary from params (3,4,3).
// One wave; lanes 0..15 each propagate basis column e_j through the circuit.
// Stores Ur (256 f32) then Ui (256 f32) row-major U[k][j] into d_ws.
// ---------------------------------------------------------------------------
__global__ __launch_bounds__(32) void build_unitary_kernel(
    const float* __restrict__ params, float* __restrict__ ws) {
  const int lane = threadIdx.x;
  float vre[DIM], vim[DIM];
#pragma unroll
  for (int k = 0; k < DIM; ++k) {
    vre[k] = (k == lane) ? 1.0f : 0.0f;
    vim[k] = 0.0f;
  }
#pragma unroll
  for (int l = 0; l < 3; ++l) {
    // Rot on each wire: Rot = RZ(omega) RY(theta) RZ(phi)
#pragma unroll
    for (int w = 0; w < N_WIRES; ++w) {
      const int pbase = (l * N_WIRES + w) * 3;
      const float phi = params[pbase + 0];
      const float th  = params[pbase + 1];
      const float om  = params[pbase + 2];
      const float al = 0.5f * (phi + om);
      const float be = 0.5f * (phi - om);
      const float ht = 0.5f * th;
      const float ca = cosf(al), sa = sinf(al);
      const float cb = cosf(be), sb = sinf(be);
      const float c = cosf(ht), s = sinf(ht);
      // r00 = ep*c, r01 = -em*s, r10 = conj(em)*s, r11 = conj(ep)*c
      const float r00r = c * ca,  r00i = -c * sa;
      const float r01r = -s * cb, r01i = -s * sb;
      const float r10r = s * cb,  r10i = -s * sb;
      const float r11r = c * ca,  r11i = c * sa;
      const int mask = 1 << (3 - w);
#pragma unroll
      for (int i0 = 0; i0 < DIM; ++i0) {
        if (i0 & mask) continue;
        const int i1 = i0 | mask;
        const float x0r = vre[i0], x0i = vim[i0];
        const float x1r = vre[i1], x1i = vim[i1];
        vre[i0] = r00r * x0r - r00i * x0i + r01r * x1r - r01i * x1i;
        vim[i0] = r00r * x0i + r00i * x0r + r01r * x1i + r01i * x1r;
        vre[i1] = r10r * x0r - r10i * x0i + r11r * x1r - r11i * x1i;
        vim[i1] = r10r * x0i + r10i * x0r + r11r * x1i + r11i * x1r;
      }
    }
    // CNOT ring, range r = l % 3 + 1
    const int rr = l % (N_WIRES - 1) + 1;
#pragma unroll
    for (int w = 0; w < N_WIRES; ++w) {
      const int c_ = w;
      const int t_ = (w + rr) & 3;
      const int mc = 1 << (3 - c_);
      const int mt = 1 << (3 - t_);
#pragma unroll
      for (int k = 0; k < DIM; ++k) {
        if ((k & mc) && !(k & mt)) {
          const int k2 = k | mt;
          float tr = vre[k]; vre[k] = vre[k2]; vre[k2] = tr;
          float ti = vim[k]; vim[k] = vim[k2]; vim[k2] = ti;
        }
      }
    }
  }
  if (lane < DIM) {
#pragma unroll
    for (int k = 0; k < DIM; ++k) {
      ws[k * DIM + lane] = vre[k];         // Ur[k][j]
      ws[256 + k * DIM + lane] = vim[k];   // Ui[k][j]
    }
  }
}

// ---------------------------------------------------------------------------
// Kernel 2: batched quanvolution via V_WMMA_F32_16X16X4_F32.
// D(16x16) = U(16x16) x Psi0^T(16x16): D lane layout puts the STATE on the
// lane axis (N = lane&15) and the component on VGPR/half (k = r + 8*hi), so
// the Z-sign reduction is compile-time adds plus one SWAPX16 ds_swizzle.
// ---------------------------------------------------------------------------
__device__ __forceinline__ float xadd16(float v) {
  // combine with lane ^ 16 (group-of-32 swizzle, xor=0x10, and=0x1f)
  return v + __int_as_float(
                 __builtin_amdgcn_ds_swizzle(__float_as_int(v), 0x401F));
}

__global__ __launch_bounds__(256) void quanv_kernel(
    const float* __restrict__ x, const float* __restrict__ calib_w,
    const float* __restrict__ calib_b, const float* __restrict__ ws,
    float* __restrict__ out) {
  const int tid = threadIdx.x;
  const int lane = tid & 31;
  const int wave = tid >> 5;
  const int m = lane & 15;   // state within tile / A-row
  const int hi = lane >> 4;  // half-wave select

  const float* Ur = ws;
  const float* Ui = ws + 256;

  // A-matrix tiles (16x4 f32): VGPR v, lane l -> U[l&15][4t + 2*hi + v]
  v2f Ar[4], Ai[4];
#pragma unroll
  for (int t = 0; t < 4; ++t) {
    const int j0 = 4 * t + 2 * hi;
    Ar[t][0] = Ur[m * 16 + j0];
    Ar[t][1] = Ur[m * 16 + j0 + 1];
    Ai[t][0] = Ui[m * 16 + j0];
    Ai[t][1] = Ui[m * 16 + j0 + 1];
  }

  float cwv[16], cbv[4];
#pragma unroll
  for (int i = 0; i < 16; ++i) cwv[i] = calib_w[i];
#pragma unroll
  for (int i = 0; i < 4; ++i) cbv[i] = calib_b[i];

  const int base = (blockIdx.x * WAVES_PER_BLOCK + wave) * TILES_PER_WAVE;

  for (int it = 0; it < TILES_PER_WAVE; ++it) {
    const int tile = base + it;
    const unsigned n = (unsigned)tile * 16u + (unsigned)m;  // global state id
    const unsigned b = n / NPATCH;
    const unsigned p = n - b * NPATCH;
    const unsigned py = p / PSIDE;
    const unsigned px = p - py * PSIDE;
    const float* src = x + (size_t)b * 784 + (size_t)py * 56 + (size_t)px * 2;
    const float2 r0 = *(const float2*)(src);       // (2py,2px), (2py,2px+1)
    const float2 r1 = *(const float2*)(src + 28);  // (2py+1,2px), (2py+1,2px+1)

    const float c0 = __cosf(0.5f * r0.x), s0 = __sinf(0.5f * r0.x);
    const float c1 = __cosf(0.5f * r0.y), s1 = __sinf(0.5f * r0.y);
    const float c2 = __cosf(0.5f * r1.x), s2 = __sinf(0.5f * r1.x);
    const float c3 = __cosf(0.5f * r1.y), s3 = __sinf(0.5f * r1.y);

    // B tile t, VGPR v, lane l -> psi0[l&15][4t + 2*hi + v]
    // j bits: [3:2]=t, [1]=hi, [0]=v ; wire0=MSB
    const float w2 = hi ? s2 : c2;
    const float pv0 = w2 * c3;
    const float pv1 = w2 * s3;
    const float g0 = c0 * c1, g1 = c0 * s1, g2 = s0 * c1, g3 = s0 * s1;
    v2f Bv[4];
    Bv[0][0] = g0 * pv0; Bv[0][1] = g0 * pv1;
    Bv[1][0] = g1 * pv0; Bv[1][1] = g1 * pv1;
    Bv[2][0] = g2 * pv0; Bv[2][1] = g2 * pv1;
    Bv[3][0] = g3 * pv0; Bv[3][1] = g3 * pv1;

    v8f Dr = {};
    v8f Di = {};
#pragma unroll
    for (int t = 0; t < 4; ++t) {
      Dr = __builtin_amdgcn_wmma_f32_16x16x4_f32(
          false, Ar[t], false, Bv[t], (short)0, Dr, false, false);
      Di = __builtin_amdgcn_wmma_f32_16x16x4_f32(
          false, Ai[t], false, Bv[t], (short)0, Di, false, false);
    }

    // probs[k = r + 8*hi] for state m, per lane
    float pr[8];
#pragma unroll
    for (int r = 0; r < 8; ++r) pr[r] = Dr[r] * Dr[r] + Di[r] * Di[r];

    // Z-signed partial sums; bit3(k)=hi, bit2..0(k)=r (compile-time signs)
    const float e01 = pr[0] + pr[1], e23 = pr[2] + pr[3];
    const float e45 = pr[4] + pr[5], e67 = pr[6] + pr[7];
    float sz0 = (e01 + e23) + (e45 + e67);
    sz0 = hi ? -sz0 : sz0;                                   // wire0 (bit3)
    const float sz1 = (e01 + e23) - (e45 + e67);             // wire1 (bit2)
    const float sz2 = (e01 + e45) - (e23 + e67);             // wire2 (bit1)
    const float sz3 = (pr[0] + pr[2] + pr[4] + pr[6]) -
                      (pr[1] + pr[3] + pr[5] + pr[7]);       // wire3 (bit0)

    const float z0 = xadd16(sz0);
    const float z1 = xadd16(sz1);
    const float z2 = xadd16(sz2);
    const float z3 = xadd16(sz3);

    if (hi == 0) {
      float4 f;
      f.x = cbv[0] + cwv[0]  * z0 + cwv[1]  * z1 + cwv[2]  * z2 + cwv[3]  * z3;
      f.y = cbv[1] + cwv[4]  * z0 + cwv[5]  * z1 + cwv[6]  * z2 + cwv[7]  * z3;
      f.z = cbv[2] + cwv[8]  * z0 + cwv[9]  * z1 + cwv[10] * z2 + cwv[11] * z3;
      f.w = cbv[3] + cwv[12] * z0 + cwv[13] * z1 + cwv[14] * z2 + cwv[15] * z3;
      *(float4*)(out + (size_t)n * 4) = f;
    }
  }
}

extern "C" void kernel_launch(void* const* d_in, const int* in_sizes, int n_in,
                              void* d_out, int out_size, void* d_ws,
                              size_t ws_size, hipStream_t stream) {
  const float* x       = (const float*)d_in[0];  // (4096,1,28,28)
  const float* params  = (const float*)d_in[1];  // (3,4,3)
  const float* calib_w = (const float*)d_in[2];  // (4,4)
  const float* calib_b = (const float*)d_in[3];  // (4,)
  float* out = (float*)d_out;                    // (4096,784)
  float* ws = (float*)d_ws;                      // Ur(256) + Ui(256) f32

  build_unitary_kernel<<<1, 32, 0, stream>>>(params, ws);
  quanv_kernel<<<NBLOCKS, 256, 0, stream>>>(x, calib_w, calib_b, ws, out);
}